// GabaLSNN_37426345017539
// MI455X (gfx1250) — compile-verified
//
#include <hip/hip_runtime.h>

#define INPUT_SZ  1024
#define HID       4096
#define OUT_SZ    512
#define T_STEPS   4096
#define LEAK      0.1f          // DT / TAU

#define SCAN_BLOCKS 128
#define SCAN_TPB    256
#define ROWS_PER_BLOCK (HID / SCAN_BLOCKS)       // 32
#define WAVES_PER_BLOCK (SCAN_TPB / 32)          // 8
#define ROWS_PER_WAVE (ROWS_PER_BLOCK / WAVES_PER_BLOCK) // 4

typedef float v2f __attribute__((ext_vector_type(2)));
typedef float v8f __attribute__((ext_vector_type(8)));

// ---------------------------------------------------------------------------
// WMMA "NT" GEMM:  C[M,N] = A[M,K] * B[N,K]^T   (K contiguous in both A and B)
// One wave computes one 16x16 tile with V_WMMA_F32_16X16X4_F32 (fp32 matches
// the fp32 reference precision). A-fragment layout (ISA 7.12.2, 32-bit A
// 16x4): lane<16 -> M=lane, K={0,1}; lane>=16 -> M=lane-16, K={2,3}; the
// B 4x16 fragment mirrors it with N in place of M. C/D: 8 VGPRs, lane<16 ->
// M=v, lane>=16 -> M=v+8, N=lane&15.
// ---------------------------------------------------------------------------
__global__ void wmma_nt_gemm(const float* __restrict__ A,
                             const float* __restrict__ B,
                             float* __restrict__ C,
                             int M, int N, int K) {
    const int wave   = (blockIdx.x * blockDim.x + threadIdx.x) >> 5;
    const int lane   = threadIdx.x & 31;
    const int tilesN = N >> 4;
    const int tileM  = (wave / tilesN) << 4;
    const int tileN  = (wave % tilesN) << 4;
    if (tileM >= M) return;                       // uniform per wave: EXEC all-1s for WMMA

    const int row  = lane & 15;                   // M index (A) / N index (B)
    const int koff = (lane >> 4) << 1;            // lane-half selects K {0,1} vs {2,3}

    const float* __restrict__ Ap = A + (size_t)(tileM + row) * K + koff;
    const float* __restrict__ Bp = B + (size_t)(tileN + row) * K + koff;

    v8f c = {};
    for (int k = 0; k < K; k += 16) {
#pragma unroll
        for (int kk = 0; kk < 16; kk += 4) {
            v2f a = *(const v2f*)(Ap + k + kk);
            v2f b = *(const v2f*)(Bp + k + kk);
            // (neg_a, A, neg_b, B, c_mod, C, reuse_a, reuse_b)
            c = __builtin_amdgcn_wmma_f32_16x16x4_f32(
                    false, a, false, b, (short)0, c, false, false);
        }
    }

    const int col   = tileN + (lane & 15);
    const int rbase = tileM + ((lane >> 4) << 3);
#pragma unroll
    for (int v = 0; v < 8; ++v)
        C[(size_t)(rbase + v) * N + col] = c[v];
}

// ---------------------------------------------------------------------------
// Init: zero h0 and the grid-barrier control words (workspace is poisoned).
// ---------------------------------------------------------------------------
__global__ void init_state(float* __restrict__ h, unsigned* __restrict__ ctrl) {
    int i = blockIdx.x * blockDim.x + threadIdx.x;
    if (i < HID) h[i] = 0.0f;
    if (i < 2)   ctrl[i] = 0u;
}

// ---------------------------------------------------------------------------
// Persistent scan kernel: 4096 sequential leaky-integrate gated steps in one
// launch. Each block owns a fixed 32-row (512 KB) slice of W_rec -> the slice
// stays hot in L2/WGP$ across all steps (W_rec is L2-resident: 64 MB of the
// 192 MB global L2). h (16 KB) is staged in LDS each step. Device-wide
// sense-free epoch barrier via agent-scope atomics with acquire/release
// fences; spinner uses s_sleep.
// ---------------------------------------------------------------------------
__global__ void scan_kernel(const float* __restrict__ Wrec,
                            const float* __restrict__ Gbias,
                            const float* __restrict__ I,      // [T, HID]
                            float* __restrict__ h,            // [HID] state
                            float* __restrict__ Hbuf,         // [T, HID]
                            unsigned* __restrict__ ctrl) {
    __shared__ float hs[HID];                     // 16 KB
    const int tid    = threadIdx.x;
    const int lane   = tid & 31;
    const int waveId = tid >> 5;
    unsigned* counter = ctrl;
    unsigned* epoch   = ctrl + 1;

    for (int t = 0; t < T_STEPS; ++t) {
        // Stage h_t into LDS (coalesced, 16 floats/thread).
        for (int i = tid; i < HID; i += SCAN_TPB) hs[i] = h[i];
        __syncthreads();

        const int rbase = blockIdx.x * ROWS_PER_BLOCK + waveId * ROWS_PER_WAVE;
#pragma unroll
        for (int rr = 0; rr < ROWS_PER_WAVE; ++rr) {
            const int r = rbase + rr;
            const float* __restrict__ wrow = Wrec + (size_t)r * HID;
            float acc = 0.0f;
            // float4-vectorized, wave-coalesced (512B/wave/iter) dot product.
#pragma unroll 4
            for (int i = 0; i < HID / (32 * 4); ++i) {
                const int idx = (i * 32 + lane) * 4;
                const float4 w  = *(const float4*)(wrow + idx);
                const float4 hv = *(const float4*)(&hs[idx]);
                acc += w.x * hv.x + w.y * hv.y + w.z * hv.z + w.w * hv.w;
            }
            // Wave32 reduction.
#pragma unroll
            for (int off = 16; off > 0; off >>= 1)
                acc += __shfl_xor(acc, off, 32);

            if (lane == 0) {
                const float rv   = acc;
                const float g    = 1.0f / (1.0f + __expf(-(Gbias[r] + rv)));
                const float z    = tanhf(I[(size_t)t * HID + r] + g * rv);
                const float hold = hs[r];
                const float hnew = hold + LEAK * (z - hold);
                h[r] = hnew;
                Hbuf[(size_t)t * HID + r] = hnew;
            }
        }

        __syncthreads();
        if (tid == 0) {
            __builtin_amdgcn_fence(__ATOMIC_RELEASE, "agent");
            const unsigned prev = __hip_atomic_fetch_add(
                counter, 1u, __ATOMIC_ACQ_REL, __HIP_MEMORY_SCOPE_AGENT);
            if (prev == SCAN_BLOCKS - 1) {
                __hip_atomic_store(counter, 0u, __ATOMIC_RELAXED,
                                   __HIP_MEMORY_SCOPE_AGENT);
                __hip_atomic_store(epoch, (unsigned)(t + 1), __ATOMIC_RELEASE,
                                   __HIP_MEMORY_SCOPE_AGENT);
            } else {
                while (__hip_atomic_load(epoch, __ATOMIC_ACQUIRE,
                                         __HIP_MEMORY_SCOPE_AGENT) <= (unsigned)t) {
                    __builtin_amdgcn_s_sleep(2);
                }
            }
            __builtin_amdgcn_fence(__ATOMIC_ACQUIRE, "agent");
        }
        __syncthreads();
    }
}

// ---------------------------------------------------------------------------
extern "C" void kernel_launch(void* const* d_in, const int* in_sizes, int n_in,
                              void* d_out, int out_size, void* d_ws, size_t ws_size,
                              hipStream_t stream) {
    const float* x      = (const float*)d_in[0];  // [T, 1024]
    const float* W_in   = (const float*)d_in[1];  // [4096, 1024]
    const float* W_rec  = (const float*)d_in[2];  // [4096, 4096]
    const float* W_out  = (const float*)d_in[3];  // [512, 4096]
    const float* G_bias = (const float*)d_in[4];  // [4096]
    float* y = (float*)d_out;                     // [T, 512]

    // Workspace layout: I (64MB) | Hbuf (64MB) | h (16KB) | ctrl
    float*    I_buf = (float*)d_ws;
    float*    Hbuf  = I_buf + (size_t)T_STEPS * HID;
    float*    h     = Hbuf + (size_t)T_STEPS * HID;
    unsigned* ctrl  = (unsigned*)(h + HID);

    // Phase 1: I = X @ W_in^T  (M=T, N=HID, K=1024) — WMMA f32
    {
        const int waves  = (T_STEPS / 16) * (HID / 16);
        const int blocks = waves / WAVES_PER_BLOCK;
        wmma_nt_gemm<<<dim3(blocks), dim3(SCAN_TPB), 0, stream>>>(
            x, W_in, I_buf, T_STEPS, HID, INPUT_SZ);
    }

    // Phase 2: zero h0 + barrier words, then one persistent scan launch.
    init_state<<<dim3((HID + 255) / 256), dim3(256), 0, stream>>>(h, ctrl);
    scan_kernel<<<dim3(SCAN_BLOCKS), dim3(SCAN_TPB), 0, stream>>>(
        W_rec, G_bias, I_buf, h, Hbuf, ctrl);

    // Phase 3: Y = H @ W_out^T  (M=T, N=512, K=4096) — WMMA f32
    {
        const int waves  = (T_STEPS / 16) * (OUT_SZ / 16);
        const int blocks = waves / WAVES_PER_BLOCK;
        wmma_nt_gemm<<<dim3(blocks), dim3(SCAN_TPB), 0, stream>>>(
            Hbuf, W_out, y, T_STEPS, OUT_SZ, HID);
    }
}